// SpiralNet_Plus_88476326298125
// MI455X (gfx1250) — compile-verified
//
#include <hip/hip_runtime.h>
#include <hip/hip_bf16.h>

// ---------------------------------------------------------------------------
// SpiralNet++ VAE forward, fp32, WMMA (v_wmma_f32_16x16x4_f32) for all GEMMs.
// Inner GEMM loop is LDS-only (A and B tiles staged), branch-free.
// ---------------------------------------------------------------------------

typedef float v2f __attribute__((ext_vector_type(2)));
typedef float v8f __attribute__((ext_vector_type(8)));

#define TILE    16
#define NWAVES  8                 // 256 threads / wave32
#define KC      64                // K features staged per chunk
#define NCOLS   (NWAVES * TILE)   // 128 output columns per block
#define BSTRIDE (NCOLS + 2)       // pad: 128 % 64banks == 0 would alias halves

// ---------------------------------------------------------------------------
// Unified gather-GEMM:  out[row, o] = act( sum_f A[row,f] * W[o,f] + bias[o] )
//   gather mode (idx != null):  row -> (b = row/nodes, n = row%nodes)
//       A[row, f] = X[b, idx[n, f/Cin], f%Cin]      (K = 9*Cin)
//   direct mode (idx == null):  A[row, f] = X[row*K + f]
// Block: 256 threads, one 16-row M tile, 128 output columns (8 waves x 16).
// ---------------------------------------------------------------------------
__global__ __launch_bounds__(256) void gemm_gather_wmma(
    const float* __restrict__ X, const int* __restrict__ idx,
    const float* __restrict__ W, const float* __restrict__ bias,
    float* __restrict__ out,
    int nodes, int Cin, int K, int Nout, int applyElu)
{
    __shared__ float ldsA[KC * TILE];     //  4 KB: [fl*16 + m]
    __shared__ float ldsB[KC * BSTRIDE];  // ~33 KB: [fl*BSTRIDE + n]

    const int tid    = threadIdx.x;
    const int lane   = tid & 31;
    const int wave   = tid >> 5;
    const int mtile  = blockIdx.x;
    const int ntiles = (Nout + TILE - 1) >> 4;
    const int ntile  = blockIdx.y * NWAVES + wave;      // wave-uniform
    const int colBase = blockIdx.y * NCOLS;
    const long rowBase = (long)mtile * TILE;

    // every 16-row tile lies inside a single batch image (nodes % 16 == 0)
    int bidx = 0, nodeBase = 0;
    if (idx) { bidx = (int)(rowBase / nodes); nodeBase = (int)(rowBase % nodes); }
    const float* __restrict__ Xb = idx ? (X + (long)bidx * nodes * Cin) : X;

    v8f acc = {0.f, 0.f, 0.f, 0.f, 0.f, 0.f, 0.f, 0.f};

    const int mloc  = lane & 15;
    const int khalf = (lane < 16) ? 0 : 2;              // K phase per half-wave
    const int nloc  = wave * TILE + mloc;               // column within ldsB

    for (int k0 = 0; k0 < K; k0 += KC) {
        __syncthreads();

        // ---- stage A chunk (gathered), zero-padded past K ----
        for (int e = tid; e < KC * TILE; e += 256) {
            const int fl = e >> 4;
            const int m  = e & 15;
            const int f  = k0 + fl;
            float v = 0.f;
            if (f < K) {
                if (idx) {
                    const int l = f / Cin;
                    const int c = f - l * Cin;
                    const int nbr = idx[(nodeBase + m) * 9 + l];
                    v = Xb[(long)nbr * Cin + c];
                } else {
                    v = X[(rowBase + m) * (long)K + f];
                }
            }
            ldsA[fl * TILE + m] = v;
        }

        // ---- stage B chunk: ldsB[k][n] = W[colBase+n][k0+k], zero-padded ----
        for (int e = tid; e < KC * NCOLS; e += 256) {
            const int n  = e >> 6;          // e / KC
            const int kl = e & (KC - 1);    // e % KC  (coalesced over k)
            const int f  = k0 + kl;
            const int col = colBase + n;
            float v = 0.f;
            if (f < K && col < Nout) v = W[(long)col * K + f];
            ldsB[kl * BSTRIDE + n] = v;
        }
        __syncthreads();

        // ---- WMMA over the chunk: pure LDS + wmma, branch-free ----
        if (ntile < ntiles) {               // wave-uniform: EXEC all-ones inside
            const int kmax = (K - k0 < KC) ? (K - k0) : KC;
            #pragma unroll 4
            for (int kk = 0; kk < kmax; kk += 4) {
                v2f a, b;
                a[0] = ldsA[(kk + khalf)     * TILE + mloc];
                a[1] = ldsA[(kk + khalf + 1) * TILE + mloc];
                b[0] = ldsB[(kk + khalf)     * BSTRIDE + nloc];
                b[1] = ldsB[(kk + khalf + 1) * BSTRIDE + nloc];
                acc = __builtin_amdgcn_wmma_f32_16x16x4_f32(
                          false, a, false, b, (short)0, acc, false, false);
            }
        }
    }

    if (ntile < ntiles) {
        const int col = ntile * TILE + mloc;
        if (col < Nout) {
            const float bv = bias[col];
            const int mhi = (lane < 16) ? 0 : 8;
            #pragma unroll
            for (int r = 0; r < 8; ++r) {
                const long row = rowBase + r + mhi;
                float v = acc[r] + bv;
                if (applyElu) v = (v > 0.f) ? v : (expf(v) - 1.f);
                out[row * (long)Nout + col] = v;
            }
        }
    }
}

// ---------------------------------------------------------------------------
// zero fill
// ---------------------------------------------------------------------------
__global__ void zero_kernel(float* __restrict__ p, long n)
{
    long i = (long)blockIdx.x * blockDim.x + threadIdx.x;
    const long stride = (long)gridDim.x * blockDim.x;
    for (; i < n; i += stride) p[i] = 0.f;
}

// ---------------------------------------------------------------------------
// COO scatter pool:  out[b, row[e], c] += X[b, col[e], c] * val[e]
// ---------------------------------------------------------------------------
__global__ void pool_scatter_kernel(
    const float* __restrict__ X, const int* __restrict__ row,
    const int* __restrict__ col, const float* __restrict__ val,
    float* __restrict__ out, int nnz, int C, int n_in, int n_out)
{
    const long t = (long)blockIdx.x * blockDim.x + threadIdx.x;
    if (t >= (long)nnz * C) return;
    const int e = (int)(t / C);
    const int c = (int)(t - (long)e * C);
    const int r  = row[e];
    const int cl = col[e];
    const float v = val[e];
    #pragma unroll
    for (int b = 0; b < 16; ++b) {
        const float g = X[((long)b * n_in + cl) * C + c] * v;
        atomicAdd(&out[((long)b * n_out + r) * C + c], g);
    }
}

// ---------------------------------------------------------------------------
// reparameterization: mu/logvar -> d_out, z = mu + eps*exp(0.5*logvar)
// lat is (16, 256): [:,0:128]=mu, [:,128:256]=logvar
// ---------------------------------------------------------------------------
__global__ void latent_kernel(
    const float* __restrict__ lat, const float* __restrict__ eps,
    float* __restrict__ z, float* __restrict__ outMu, float* __restrict__ outLv)
{
    const int t = blockIdx.x * blockDim.x + threadIdx.x;
    if (t >= 16 * 128) return;
    const int b = t >> 7;
    const int j = t & 127;
    const float mu = lat[b * 256 + j];
    const float lv = lat[b * 256 + 128 + j];
    z[t]     = mu + eps[t] * expf(0.5f * lv);
    outMu[t] = mu;
    outLv[t] = lv;
}

// ---------------------------------------------------------------------------
// host-side helpers
// ---------------------------------------------------------------------------
static inline void launch_gemm(const float* X, const int* idx, const float* W,
                               const float* b, float* out, int M, int nodes,
                               int Cin, int K, int Nout, int elu, hipStream_t s)
{
    const int ntiles = (Nout + TILE - 1) / TILE;
    dim3 grid(M / TILE, (ntiles + NWAVES - 1) / NWAVES);
    gemm_gather_wmma<<<grid, 256, 0, s>>>(X, idx, W, b, out, nodes, Cin, K, Nout, elu);
}

static inline void launch_pool(const float* X, const int* r, const int* c,
                               const float* v, float* out, int nnz, int C,
                               int n_in, int n_out, hipStream_t s)
{
    const long zn = 16L * n_out * C;
    zero_kernel<<<1024, 256, 0, s>>>(out, zn);
    const long total = (long)nnz * C;
    pool_scatter_kernel<<<(int)((total + 255) / 256), 256, 0, s>>>(
        X, r, c, v, out, nnz, C, n_in, n_out);
}

extern "C" void kernel_launch(void* const* d_in, const int* in_sizes, int n_in,
                              void* d_out, int out_size, void* d_ws, size_t ws_size,
                              hipStream_t stream)
{
    static const int N[5]  = {16384, 4096, 1024, 256, 64};
    static const int CH[5] = {3, 32, 64, 128, 256};

    const float* x   = (const float*)d_in[0];
    const float* eps = (const float*)d_in[1];
    const int* idx[4];
    for (int i = 0; i < 4; ++i) idx[i] = (const int*)d_in[2 + i];

    const int   *rd[4], *cd[4], *ru[4], *cu[4];
    const float *vd[4], *vu[4];
    for (int i = 0; i < 4; ++i) {
        rd[i] = (const int*)  d_in[6  + 3 * i];
        cd[i] = (const int*)  d_in[7  + 3 * i];
        vd[i] = (const float*)d_in[8  + 3 * i];
        ru[i] = (const int*)  d_in[18 + 3 * i];
        cu[i] = (const int*)  d_in[19 + 3 * i];
        vu[i] = (const float*)d_in[20 + 3 * i];
    }
    const float *We[4], *be[4];
    for (int i = 0; i < 4; ++i) {
        We[i] = (const float*)d_in[30 + 2 * i];
        be[i] = (const float*)d_in[31 + 2 * i];
    }
    const float* Wel = (const float*)d_in[38];
    const float* bel = (const float*)d_in[39];
    const float* Wdl = (const float*)d_in[40];
    const float* bdl = (const float*)d_in[41];
    const float *Wd[4], *bd[4];
    for (int i = 0; i < 4; ++i) {
        Wd[i] = (const float*)d_in[42 + 2 * i];
        bd[i] = (const float*)d_in[43 + 2 * i];
    }
    const float* Wdf = (const float*)d_in[50];
    const float* bdf = (const float*)d_in[51];

    // workspace layout (floats)
    float* ws = (float*)d_ws;
    const long BUF0 = 16L * 16384 * 64;   // 16,777,216 floats (largest unpool)
    const long BUF1 = 16L * 16384 * 32;   //  8,388,608 floats (largest conv out)
    float* buf0 = ws;
    float* buf1 = ws + BUF0;
    float* lat  = ws + BUF0 + BUF1;       // 16*256
    float* zbuf = lat + 16 * 256;         // 16*128

    float* re_x   = (float*)d_out;                        // 16*16384*3
    float* out_mu = re_x + 16L * 16384 * 3;               // 16*128
    float* out_lv = out_mu + 16 * 128;                    // 16*128

    // ---------------- encoder ----------------
    launch_gemm(x, idx[0], We[0], be[0], buf1, 16 * N[0], N[0], CH[0], 9 * CH[0], CH[1], 1, stream);
    launch_pool(buf1, rd[0], cd[0], vd[0], buf0, 3 * N[1], CH[1], N[0], N[1], stream);
    launch_gemm(buf0, idx[1], We[1], be[1], buf1, 16 * N[1], N[1], CH[1], 9 * CH[1], CH[2], 1, stream);
    launch_pool(buf1, rd[1], cd[1], vd[1], buf0, 3 * N[2], CH[2], N[1], N[2], stream);
    launch_gemm(buf0, idx[2], We[2], be[2], buf1, 16 * N[2], N[2], CH[2], 9 * CH[2], CH[3], 1, stream);
    launch_pool(buf1, rd[2], cd[2], vd[2], buf0, 3 * N[3], CH[3], N[2], N[3], stream);
    launch_gemm(buf0, idx[3], We[3], be[3], buf1, 16 * N[3], N[3], CH[3], 9 * CH[3], CH[4], 1, stream);
    launch_pool(buf1, rd[3], cd[3], vd[3], buf0, 3 * N[4], CH[4], N[3], N[4], stream);

    // ---------------- latent ----------------
    // lat = h @ Wel.T + bel   (M=16, K=16384, N=256)
    launch_gemm(buf0, nullptr, Wel, bel, lat, 16, 1, 0, N[4] * CH[4], 2 * 128, 0, stream);
    latent_kernel<<<(16 * 128 + 255) / 256, 256, 0, stream>>>(lat, eps, zbuf, out_mu, out_lv);
    // hd = z @ Wdl.T + bdl    (M=16, K=128, N=16384) -> (16, 64, 256)
    launch_gemm(zbuf, nullptr, Wdl, bdl, buf1, 16, 1, 0, 128, N[4] * CH[4], 0, stream);

    // ---------------- decoder ----------------
    launch_pool(buf1, ru[3], cu[3], vu[3], buf0, 3 * N[3], CH[4], N[4], N[3], stream);
    launch_gemm(buf0, idx[3], Wd[0], bd[0], buf1, 16 * N[3], N[3], 256, 9 * 256, 256, 1, stream);
    launch_pool(buf1, ru[2], cu[2], vu[2], buf0, 3 * N[2], 256, N[3], N[2], stream);
    launch_gemm(buf0, idx[2], Wd[1], bd[1], buf1, 16 * N[2], N[2], 256, 9 * 256, 128, 1, stream);
    launch_pool(buf1, ru[1], cu[1], vu[1], buf0, 3 * N[1], 128, N[2], N[1], stream);
    launch_gemm(buf0, idx[1], Wd[2], bd[2], buf1, 16 * N[1], N[1], 128, 9 * 128, 64, 1, stream);
    launch_pool(buf1, ru[0], cu[0], vu[0], buf0, 3 * N[0], 64, N[1], N[0], stream);
    launch_gemm(buf0, idx[0], Wd[3], bd[3], buf1, 16 * N[0], N[0], 64, 9 * 64, 32, 1, stream);
    // final conv Wdf (3 <- 32*9), no ELU, straight into d_out
    launch_gemm(buf1, idx[0], Wdf, bdf, re_x, 16 * N[0], N[0], 32, 9 * 32, 3, 0, stream);
}